// buildProposalTargetLayer_15899968930080
// MI455X (gfx1250) — compile-verified
//
#include <hip/hip_runtime.h>
#include <hip/hip_bf16.h>
#include <stdint.h>

// ---------------------------------------------------------------------------
// Proposal-target layer for MI455X (gfx1250).
// IoU is min/max/clip math -> no bilinear form -> WMMA not applicable.
// CDNA5 data path used instead: async global->LDS staging of GT boxes
// (global_load_async_to_lds_b128 / s_wait_asynccnt), global_prefetch_b8,
// wave32 ballot-based deterministic stream compaction.
// ---------------------------------------------------------------------------

#define MAXG 128           // cap on GT boxes held in LDS (reference uses 100)
#define CHUNK 256          // rois per block in kernels 1 and 3

// ---- CDNA5 async global->LDS helpers (inline asm: portable across both
// ROCm-7.2 and amdgpu-toolchain, which disagree on the builtin arity).
__device__ __forceinline__ void async_g2lds_b128(unsigned lds_byte_off,
                                                 unsigned vgpr_byte_off,
                                                 const void* sbase) {
  unsigned long long base = (unsigned long long)sbase;
  asm volatile("global_load_async_to_lds_b128 %0, %1, %2"
               :: "v"(lds_byte_off), "v"(vgpr_byte_off), "s"(base)
               : "memory");
}
__device__ __forceinline__ void async_g2lds_b32(unsigned lds_byte_off,
                                                unsigned vgpr_byte_off,
                                                const void* sbase) {
  unsigned long long base = (unsigned long long)sbase;
  asm volatile("global_load_async_to_lds_b32 %0, %1, %2"
               :: "v"(lds_byte_off), "v"(vgpr_byte_off), "s"(base)
               : "memory");
}
__device__ __forceinline__ void wait_asynccnt0() {
  asm volatile("s_wait_asynccnt 0x0" ::: "memory");
}

// ---- small deterministic hash RNG (reference RNG is JAX threefry; the
// sampling stage has no bit-exact spec to hit, only the same structure).
__device__ __forceinline__ unsigned hash_u32(unsigned x) {
  x ^= x >> 17; x *= 0xed5ad4bbu;
  x ^= x >> 11; x *= 0xac4c1b51u;
  x ^= x >> 15; x *= 0x31848babu;
  x ^= x >> 14;
  return x;
}
__device__ __forceinline__ float u01(unsigned h) {
  return (float)(h >> 8) * (1.0f / 16777216.0f);   // [0,1)
}

// ---------------------------------------------------------------------------
// Kernel 1: IoU max/argmax + fg/bg classification + per-chunk counts.
// grid = (CH, B), block = 256.
//   assign[b*N+n] = argmax_g | (fg ? 0x80000000 : 0)
//   cfg/cbg[b*CH+chunk] = #fg / #bg rois in this chunk
// Inner loop is division-free: track (bestNum,bestDen) and compare by
// cross-multiplication (denominators strictly positive).
// ---------------------------------------------------------------------------
__global__ void k_iou(const float* __restrict__ all_rois,
                      const float* __restrict__ gt,
                      int R, int G, int N, int CH,
                      int* __restrict__ assign,
                      int* __restrict__ cfg, int* __restrict__ cbg) {
  __shared__ __align__(16) float sRaw[MAXG * 5];   // async staging target
  __shared__ float sX1[MAXG], sY1[MAXG], sX2[MAXG], sY2[MAXG], sAr[MAXG];
  __shared__ int sCnt[2];
  const int b   = blockIdx.y;
  const int tid = threadIdx.x;
  const int Gc  = (G < MAXG) ? G : MAXG;

  // ---- async stage this image's GT slice (G*5 floats) into LDS ----
  const int total = Gc * 5;                          // dwords
  const float* gsrc = gt + (size_t)b * G * 5;
  const unsigned ldsbase = (unsigned)(size_t)(&sRaw[0]);
  if ((total & 3) == 0 && (((size_t)gsrc) & 15) == 0) {
    const int nvec = total >> 2;                     // 16B packets
    for (int i = tid; i < nvec; i += blockDim.x)
      async_g2lds_b128(ldsbase + (unsigned)i * 16u, (unsigned)i * 16u, gsrc);
  } else {
    for (int i = tid; i < total; i += blockDim.x)
      async_g2lds_b32(ldsbase + (unsigned)i * 4u, (unsigned)i * 4u, gsrc);
  }
  if (tid < 2) sCnt[tid] = 0;

  // ---- load my ROI while the async copy is in flight ----
  const int n = blockIdx.x * CHUNK + tid;
  const bool valid = (n < N);
  float x1 = 0.f, y1 = 0.f, x2 = 0.f, y2 = 0.f;
  if (valid) {
    if (n < R) {
      const float* p = all_rois + ((size_t)b * R + n) * 5;
      __builtin_prefetch(p, 0, 0);                   // global_prefetch_b8
      x1 = p[1]; y1 = p[2]; x2 = p[3]; y2 = p[4];
    } else {                                         // appended GT rois
      const float* p = gt + ((size_t)b * G + (n - R)) * 5;
      x1 = p[0]; y1 = p[1]; x2 = p[2]; y2 = p[3];
    }
  }

  wait_asynccnt0();        // my wave's async loads landed in LDS
  __syncthreads();         // everyone's did

  // ---- AoS -> SoA + precompute gt areas (once per block) ----
  if (tid < Gc) {
    const float gx1 = sRaw[tid * 5 + 0];
    const float gy1 = sRaw[tid * 5 + 1];
    const float gx2 = sRaw[tid * 5 + 2];
    const float gy2 = sRaw[tid * 5 + 3];
    sX1[tid] = gx1; sY1[tid] = gy1; sX2[tid] = gx2; sY2[tid] = gy2;
    sAr[tid] = (gx2 - gx1 + 1.0f) * (gy2 - gy1 + 1.0f);
  }
  __syncthreads();

  // ---- IoU max / argmax (first-max tie-break, like jnp.argmax) ----
  const float areaA = (x2 - x1 + 1.0f) * (y2 - y1 + 1.0f);
  float bestNum = -1.0f;   // best IoU numerator   (inter)
  float bestDen =  1.0f;   // best IoU denominator (union, > 0)
  int   bi      = 0;
  for (int g = 0; g < Gc; ++g) {
    // all lanes read the same LDS address -> conflict-free broadcast
    const float gx1 = sX1[g], gy1 = sY1[g], gx2 = sX2[g], gy2 = sY2[g];
    float iw = fminf(x2, gx2) - fmaxf(x1, gx1) + 1.0f;
    float ih = fminf(y2, gy2) - fmaxf(y1, gy1) + 1.0f;
    iw = fmaxf(iw, 0.0f);
    ih = fmaxf(ih, 0.0f);
    const float inter = iw * ih;
    const float den   = areaA + sAr[g] - inter;      // >= max(areaA,areaB) > 0
    // inter/den > bestNum/bestDen  <=>  inter*bestDen > bestNum*den
    if (inter * bestDen > bestNum * den) { bestNum = inter; bestDen = den; bi = g; }
  }
  const float best = bestNum / bestDen;              // one divide, after loop

  const bool fg = valid && (best >= 0.5f);           // FG_THRESH
  const bool bg = valid && !fg;                      // IoU in [0,1] => bg range
  if (valid) {
    unsigned a = (unsigned)bi | (fg ? 0x80000000u : 0u);
    assign[(size_t)b * N + n] = (int)a;
  }
  if (fg) atomicAdd(&sCnt[0], 1);
  if (bg) atomicAdd(&sCnt[1], 1);
  __syncthreads();
  if (tid == 0) {
    cfg[(size_t)b * CH + blockIdx.x] = sCnt[0];
    cbg[(size_t)b * CH + blockIdx.x] = sCnt[1];
  }
}

// ---------------------------------------------------------------------------
// Kernel 2: per-image exclusive scan of chunk counts (CH ~ 79, trivial).
// grid = B, block = 32.
// ---------------------------------------------------------------------------
__global__ void k_scan(const int* __restrict__ cfg, const int* __restrict__ cbg,
                       int* __restrict__ ofg, int* __restrict__ obg,
                       int* __restrict__ cnt, int CH) {
  const int b = blockIdx.x;
  if (threadIdx.x == 0) {
    int s = 0;
    for (int c = 0; c < CH; ++c) { ofg[(size_t)b * CH + c] = s; s += cfg[(size_t)b * CH + c]; }
    cnt[2 * b] = s;
    s = 0;
    for (int c = 0; c < CH; ++c) { obg[(size_t)b * CH + c] = s; s += cbg[(size_t)b * CH + c]; }
    cnt[2 * b + 1] = s;
  }
}

// ---------------------------------------------------------------------------
// Kernel 3: deterministic compaction of fg / bg roi indices.
// grid = (CH, B), block = 256 (8 wave32s). Rank = ballot prefix within wave
// + LDS scan over waves + chunk offset -> replay-deterministic ordering.
// ---------------------------------------------------------------------------
__global__ void k_compact(const int* __restrict__ assign,
                          const int* __restrict__ ofg, const int* __restrict__ obg,
                          int N, int CH,
                          int* __restrict__ fgidx, int* __restrict__ bgidx) {
  __shared__ int wF[8], wB[8];
  const int b    = blockIdx.y;
  const int tid  = threadIdx.x;
  const int lane = tid & 31;           // wave32 (gfx1250 is wave32-only)
  const int w    = tid >> 5;
  const int n    = blockIdx.x * CHUNK + tid;
  const bool valid = (n < N);
  int a = 0;
  if (valid) a = assign[(size_t)b * N + n];
  const bool fg = valid && ((unsigned)a & 0x80000000u);
  const bool bg = valid && !fg;

  const unsigned mF = (unsigned)__ballot(fg);
  const unsigned mB = (unsigned)__ballot(bg);
  if (lane == 0) { wF[w] = __popc(mF); wB[w] = __popc(mB); }
  __syncthreads();
  if (tid == 0) {
    int s = 0;
    for (int i = 0; i < 8; ++i) { int t = wF[i]; wF[i] = s; s += t; }
    s = 0;
    for (int i = 0; i < 8; ++i) { int t = wB[i]; wB[i] = s; s += t; }
  }
  __syncthreads();
  const unsigned lmask = (lane == 0) ? 0u : (0xffffffffu >> (32 - lane));
  if (fg) {
    const int pos = ofg[(size_t)b * CH + blockIdx.x] + wF[w] + __popc(mF & lmask);
    fgidx[(size_t)b * N + pos] = n;
  }
  if (bg) {
    const int pos = obg[(size_t)b * CH + blockIdx.x] + wB[w] + __popc(mB & lmask);
    bgidx[(size_t)b * N + pos] = n;
  }
}

// ---------------------------------------------------------------------------
// Kernel 4: sample S rois per image, gather, encode bbox targets, write out.
// grid = B, block = S (512).
// ---------------------------------------------------------------------------
__global__ void k_sample(const float* __restrict__ all_rois,
                         const float* __restrict__ gt,
                         const float* __restrict__ means,
                         const float* __restrict__ stds,
                         const int* __restrict__ assign,
                         const int* __restrict__ fgidx,
                         const int* __restrict__ bgidx,
                         const int* __restrict__ cnt,
                         int R, int G, int N,
                         float* __restrict__ out_rois,
                         float* __restrict__ out_labels,
                         float* __restrict__ out_targets) {
  const int b = blockIdx.x;
  const int s = threadIdx.x;
  const int S = blockDim.x;
  const int nfg = cnt[2 * b];
  const int nbg = cnt[2 * b + 1];
  const int FG_PER_IMAGE = S / 4;                       // round(0.25*S)

  int n_fg_take = (nbg > 0) ? min(FG_PER_IMAGE, nfg) : S;
  if (nfg == 0) n_fg_take = 0;

  const unsigned h  = hash_u32((unsigned)b * 0x9e3779b9u ^ (unsigned)s * 0x85ebca6bu ^ 0x2a2au);
  const unsigned hb = hash_u32((unsigned)b ^ 0xc0ffee42u); // per-image rotation
  const bool is_fg = (s < n_fg_take);

  int pick;
  if (is_fg) {
    int idx;
    if (nbg > 0) {
      // without replacement: rotated walk over the packed fg list (s < nfg here)
      idx = (int)(((unsigned)s + hb % (unsigned)nfg) % (unsigned)nfg);
    } else {
      // no-bg path: fg with replacement
      idx = min((int)(u01(h) * (float)nfg), nfg - 1);
    }
    pick = fgidx[(size_t)b * N + idx];
  } else {
    const int idx = (nbg > 0) ? min((int)(u01(h) * (float)nbg), nbg - 1) : 0;
    pick = bgidx[(size_t)b * N + idx];
  }

  // gather roi box
  float r1, r2, r3, r4;
  if (pick < R) {
    const float* p = all_rois + ((size_t)b * R + pick) * 5;
    r1 = p[1]; r2 = p[2]; r3 = p[3]; r4 = p[4];
  } else {
    const float* p = gt + ((size_t)b * G + (pick - R)) * 5;
    r1 = p[0]; r2 = p[1]; r3 = p[2]; r4 = p[3];
  }
  // gather assigned gt box + label
  const int ga = (int)((unsigned)assign[(size_t)b * N + pick] & 0x7fffffffu);
  const float* q = gt + ((size_t)b * G + ga) * 5;
  const float g1 = q[0], g2 = q[1], g3 = q[2], g4 = q[3];
  const float label = is_fg ? q[4] : 0.0f;

  // rois output (col 0 = image index, as in reference .at[:,0].set(img_idx))
  float* ro = out_rois + ((size_t)b * S + s) * 5;
  ro[0] = (float)b; ro[1] = r1; ro[2] = r2; ro[3] = r3; ro[4] = r4;
  out_labels[(size_t)b * S + s] = label;

  // encode (dx, dy, log dw, log dh)
  const float ew = r3 - r1 + 1.0f, eh = r4 - r2 + 1.0f;
  const float ecx = r1 + 0.5f * ew, ecy = r2 + 0.5f * eh;
  const float gw = g3 - g1 + 1.0f, gh = g4 - g2 + 1.0f;
  const float gcx = g1 + 0.5f * gw, gcy = g2 + 0.5f * gh;
  float t0 = (gcx - ecx) / ew;
  float t1 = (gcy - ecy) / eh;
  float t2 = __logf(gw / ew);
  float t3 = __logf(gh / eh);
  const float msk = (label > 0.0f) ? 1.0f : 0.0f;
  t0 = (t0 - means[0]) / stds[0] * msk;
  t1 = (t1 - means[1]) / stds[1] * msk;
  t2 = (t2 - means[2]) / stds[2] * msk;
  t3 = (t3 - means[3]) / stds[3] * msk;
  float* to = out_targets + ((size_t)b * S + s) * 4;
  to[0] = t0; to[1] = t1; to[2] = t2; to[3] = t3;
}

// ---------------------------------------------------------------------------
extern "C" void kernel_launch(void* const* d_in, const int* in_sizes, int n_in,
                              void* d_out, int out_size, void* d_ws, size_t ws_size,
                              hipStream_t stream) {
  const float* all_rois = (const float*)d_in[0];   // [B,R,5]
  const float* gt       = (const float*)d_in[1];   // [B,G,5]
  const float* means    = (const float*)d_in[2];   // [4]
  const float* stds     = (const float*)d_in[3];   // [4]
  // d_in[4] (num_gt_boxes) accepted but unused, as in the reference.

  const int B = in_sizes[4];
  const int G = in_sizes[1] / (5 * B);
  const int R = in_sizes[0] / (5 * B);
  const int N = R + G;
  const int CH = (N + CHUNK - 1) / CHUNK;
  const int S = out_size / (10 * B);               // 512 for the reference shapes

  // workspace carve-up (ints): ~3.9 MB for reference shapes
  int* assign = (int*)d_ws;                        // B*N
  int* fgidx  = assign + (size_t)B * N;            // B*N
  int* bgidx  = fgidx  + (size_t)B * N;            // B*N
  int* cfg    = bgidx  + (size_t)B * N;            // B*CH
  int* cbg    = cfg    + (size_t)B * CH;           // B*CH
  int* ofg    = cbg    + (size_t)B * CH;           // B*CH
  int* obg    = ofg    + (size_t)B * CH;           // B*CH
  int* cnt    = obg    + (size_t)B * CH;           // 2*B

  float* out_rois    = (float*)d_out;              // [B,S,5]
  float* out_labels  = out_rois   + (size_t)B * S * 5;   // [B,S]
  float* out_targets = out_labels + (size_t)B * S;       // [B,S,4]

  k_iou<<<dim3(CH, B), CHUNK, 0, stream>>>(all_rois, gt, R, G, N, CH,
                                           assign, cfg, cbg);
  k_scan<<<B, 32, 0, stream>>>(cfg, cbg, ofg, obg, cnt, CH);
  k_compact<<<dim3(CH, B), CHUNK, 0, stream>>>(assign, ofg, obg, N, CH,
                                               fgidx, bgidx);
  k_sample<<<B, S, 0, stream>>>(all_rois, gt, means, stds,
                                assign, fgidx, bgidx, cnt, R, G, N,
                                out_rois, out_labels, out_targets);
}